// Quantizer_58626303590828
// MI455X (gfx1250) — compile-verified
//
#include <hip/hip_runtime.h>
#include <hip/hip_bf16.h>

typedef __attribute__((ext_vector_type(2))) float v2f;
typedef __attribute__((ext_vector_type(8))) float v8f;

#define NUM_EMB   512
#define EMB_DIM   64
#define N_VEC     131072            // 32 * 64 * 64
#define N_TILES   (N_VEC / 16)      // 8192
#define HWSZ      4096              // 64*64 spatial per batch
#define BSTRIDE   262144            // 64 * 4096 floats per batch in [B,C,H,W]

// output layout (floats), tuple flattened in return order
#define OFF_LOSS   0
#define OFF_QUANT  1
#define QUANT_SZ   8388608          // 32*64*64*64
#define OFF_PERP   (OFF_QUANT + QUANT_SZ)              // 8388609
#define OFF_ENC    (OFF_PERP + 1)                      // 8388610
#define ENC_SZ     (N_VEC * NUM_EMB)                   // 67108864
#define OFF_IDX    (OFF_ENC + ENC_SZ)                  // 75497474

// ---------------------------------------------------------------------------
// Kernel 1: per-code squared norms + zero the usage histogram
// ---------------------------------------------------------------------------
__global__ void prep_kernel(const float* __restrict__ emb,
                            float* __restrict__ enorm,
                            int* __restrict__ counts) {
    int c = blockIdx.x * blockDim.x + threadIdx.x;
    if (c < NUM_EMB) {
        const float4* e4 = (const float4*)(emb + (size_t)c * EMB_DIM);
        float s = 0.f;
        #pragma unroll
        for (int i = 0; i < EMB_DIM / 4; ++i) {
            float4 v = e4[i];
            s += v.x * v.x + v.y * v.y + v.z * v.z + v.w * v.w;
        }
        enorm[c] = s;
        counts[c] = 0;
    }
}

// ---------------------------------------------------------------------------
// Kernel 2: WMMA distance GEMM + argmin. One wave32 per 16-row tile.
//   B operand (4x16 f32): lanes 0-15 -> N=lane, VGPR j holds K = j;
//                         lanes 16-31 -> N=lane-16, VGPR j holds K = 2+j.
//   A operand (16x4 f32): lanes 0-15 -> M=lane, K = j; lanes 16-31 -> K = 2+j.
//   D (16x16 f32): lanes 0-15 -> N=lane, rows M=r; lanes 16-31 -> rows M=8+r.
// D = z . e (raw); dist(-||z||^2) = fma(-2, D, ||e||^2); ||z||^2 added only
// for the loss (argmin is invariant to it).
// ---------------------------------------------------------------------------
__global__ void argmin_wmma_kernel(const float* __restrict__ z,
                                   const float* __restrict__ emb,
                                   const float* __restrict__ enorm,
                                   int* __restrict__ ws_idx,
                                   float* __restrict__ ws_tilesum,
                                   int* __restrict__ ws_counts,
                                   float* __restrict__ out) {
    const int wavesPerBlock = blockDim.x >> 5;
    const int T = blockIdx.x * wavesPerBlock + (threadIdx.x >> 5);
    if (T >= N_TILES) return;
    const int lane = threadIdx.x & 31;
    const int nl   = lane & 15;       // local row (z vector) within tile
    const int hi   = lane >> 4;       // half-wave selector
    const int khalf = hi * 2;         // K sub-offset within a WMMA step

    const int n0 = T * 16;            // flat z row base (same batch: 4096 % 16 == 0)
    const int b  = n0 >> 12;
    const int s0 = n0 & (HWSZ - 1);
    const float* zb = z + (size_t)b * BSTRIDE + s0 + nl;

    // Preload full 16x64 z tile into B-operand layout (raw values).
    v2f zr[16];
    float zsq = 0.f;
    #pragma unroll
    for (int kk = 0; kk < 16; ++kk) {
        float a0 = zb[(size_t)(4 * kk + khalf + 0) * HWSZ];
        float a1 = zb[(size_t)(4 * kk + khalf + 1) * HWSZ];
        zsq += a0 * a0 + a1 * a1;     // this lane covers half the dims of row nl
        zr[kk][0] = a0;
        zr[kk][1] = a1;
    }

    float minv = 3.4e38f;
    int   minc = 0;
    #pragma unroll 1
    for (int ct = 0; ct < NUM_EMB / 16; ++ct) {
        // Preload 16 A operands (code tile, all K steps) as b64 loads.
        v2f av[16];
        const float* ebase = emb + (size_t)(ct * 16 + nl) * EMB_DIM + khalf;
        #pragma unroll
        for (int kk = 0; kk < 16; ++kk)
            av[kk] = *(const v2f*)(ebase + 4 * kk);

        v8f acc = {};
        #pragma unroll
        for (int kk = 0; kk < 16; ++kk)
            acc = __builtin_amdgcn_wmma_f32_16x16x4_f32(
                false, av[kk], false, zr[kk], (short)0, acc, false, false);

        const int cbase = ct * 16 + hi * 8;   // codes this lane's D rows cover
        const float4* en4 = (const float4*)(enorm + cbase);
        float4 e0 = en4[0], e1 = en4[1];
        float en[8] = {e0.x, e0.y, e0.z, e0.w, e1.x, e1.y, e1.z, e1.w};
        #pragma unroll
        for (int r = 0; r < 8; ++r) {
            float d = fmaf(-2.f, acc[r], en[r]); // ||e||^2 - 2 z.e
            if (d < minv) { minv = d; minc = cbase + r; }   // strict < => first min
        }
    }

    // Combine the two half-wave row partitions (codes 0-7 vs 8-15 mod 16).
    float ov   = __shfl_xor(minv, 16, 32);
    int   oc   = __shfl_xor(minc, 16, 32);
    float ozsq = __shfl_xor(zsq, 16, 32);
    if (ov < minv || (ov == minv && oc < minc)) { minv = ov; minc = oc; }
    const float full = minv + zsq + ozsq;     // = ||z - e_min||^2

    // Tile partial loss (lanes 0-15 and 16-31 hold identical values; reduce one half).
    float s = full;
    s += __shfl_xor(s, 1, 32);
    s += __shfl_xor(s, 2, 32);
    s += __shfl_xor(s, 4, 32);
    s += __shfl_xor(s, 8, 32);
    if (lane == 0) ws_tilesum[T] = s;

    if (lane < 16) {
        const int n = n0 + nl;
        ws_idx[n] = minc;
        out[OFF_IDX + (size_t)n] = (float)minc;
        atomicAdd(&ws_counts[minc], 1);       // integer: deterministic
    }
}

// ---------------------------------------------------------------------------
// Kernel 3: one-hot encodings, float4 coalesced stores (dominant traffic)
// ---------------------------------------------------------------------------
__global__ void encodings_kernel(const int* __restrict__ ws_idx,
                                 float* __restrict__ out) {
    const size_t t = (size_t)blockIdx.x * blockDim.x + threadIdx.x; // N*128 float4s
    const int n  = (int)(t >> 7);
    const int c4 = ((int)t & 127) << 2;
    const int id = ws_idx[n];
    float4 v = make_float4(0.f, 0.f, 0.f, 0.f);
    const int j = id - c4;
    if (j >= 0 && j < 4) (&v.x)[j] = 1.f;
    ((float4*)(out + OFF_ENC))[t] = v;
}

// ---------------------------------------------------------------------------
// Kernel 4: quantized output in [B, C, H, W] layout, output-linear mapping
// ---------------------------------------------------------------------------
__global__ void quantized_kernel(const int* __restrict__ ws_idx,
                                 const float* __restrict__ emb,
                                 float* __restrict__ out) {
    const size_t t = (size_t)blockIdx.x * blockDim.x + threadIdx.x; // 8388608
    const int s = (int)t & (HWSZ - 1);
    const int d = ((int)(t >> 12)) & (EMB_DIM - 1);
    const int b = (int)(t >> 18);
    const int n = (b << 12) + s;
    const int id = ws_idx[n];
    out[OFF_QUANT + t] = emb[(size_t)id * EMB_DIM + d];
}

// ---------------------------------------------------------------------------
// Kernel 5: scalar reductions (single block, fixed-order: deterministic)
// ---------------------------------------------------------------------------
__global__ void finalize_kernel(const float* __restrict__ ws_tilesum,
                                const int* __restrict__ ws_counts,
                                float* __restrict__ out) {
    __shared__ float red[512];
    const int tid = threadIdx.x;

    float ls = 0.f;
    for (int i = tid; i < N_TILES; i += 512) ls += ws_tilesum[i];
    red[tid] = ls;
    __syncthreads();
    for (int ofs = 256; ofs > 0; ofs >>= 1) {
        if (tid < ofs) red[tid] += red[tid + ofs];
        __syncthreads();
    }
    const float total_loss = red[0];
    __syncthreads();

    {
        const float p = (float)ws_counts[tid] * (1.f / (float)N_VEC);
        red[tid] = p * __logf(p + 1e-10f);
    }
    __syncthreads();
    for (int ofs = 256; ofs > 0; ofs >>= 1) {
        if (tid < ofs) red[tid] += red[tid + ofs];
        __syncthreads();
    }
    if (tid == 0) {
        out[OFF_LOSS] = 1.25f * total_loss / (float)QUANT_SZ; // (1 + BETA) * mse
        out[OFF_PERP] = __expf(-red[0]);
    }
}

// ---------------------------------------------------------------------------
extern "C" void kernel_launch(void* const* d_in, const int* in_sizes, int n_in,
                              void* d_out, int out_size, void* d_ws, size_t ws_size,
                              hipStream_t stream) {
    const float* z   = (const float*)d_in[0];   // [32, 64, 64, 64]
    const float* emb = (const float*)d_in[1];   // [512, 64]
    float* out = (float*)d_out;

    float* enorm   = (float*)d_ws;              // 512
    int*   counts  = (int*)(enorm + NUM_EMB);   // 512
    float* tilesum = (float*)(counts + NUM_EMB);// 8192
    int*   idxws   = (int*)(tilesum + N_TILES); // 131072

    prep_kernel<<<(NUM_EMB + 255) / 256, 256, 0, stream>>>(emb, enorm, counts);

    // 8192 tiles, 8 waves per 256-thread block
    argmin_wmma_kernel<<<N_TILES / 8, 256, 0, stream>>>(
        z, emb, enorm, idxws, tilesum, counts, out);

    encodings_kernel<<<(N_VEC * (NUM_EMB / 4)) / 256, 256, 0, stream>>>(idxws, out);

    quantized_kernel<<<QUANT_SZ / 256, 256, 0, stream>>>(idxws, emb, out);

    finalize_kernel<<<1, 512, 0, stream>>>(tilesum, counts, out);
}